// HyperbolicMemoryLayer_28707561407111
// MI455X (gfx1250) — compile-verified
//
#include <hip/hip_runtime.h>
#include <hip/hip_bf16.h>
#include <math.h>

typedef __attribute__((ext_vector_type(2))) float v2f;
typedef __attribute__((ext_vector_type(4))) float v4f;
typedef __attribute__((ext_vector_type(8))) float v8f;

#define DIM 256
#define BQ  64
#define TOPK 32

// Use inline CDNA5 asm for the async global->LDS copy path (the clang builtin
// has a HIP-unspellable pointer prototype; asm bypasses it per the ISA doc).
#define USE_ASYNC_ASM 1

#if USE_ASYNC_ASM
__device__ __forceinline__ void async_ld_b128(const float* g, float* l) {
    // VDST = 32-bit workgroup-relative LDS byte offset (low 32 bits of the
    // generic shared pointer); VADDR = 64-bit global address; no SADDR.
    unsigned int loff = (unsigned int)(unsigned long long)l;
    asm volatile("global_load_async_to_lds_b128 %0, %1, off"
                 :: "v"(loff), "v"(g) : "memory");
}
__device__ __forceinline__ void wait_async_le6() {
    asm volatile("s_wait_asynccnt 0x6" ::: "memory");
}
__device__ __forceinline__ void wait_async_0() {
    asm volatile("s_wait_asynccnt 0x0" ::: "memory");
}
#endif

// Stage one 16-byte quantum: async copy engine (CDNA5) or sync fallback.
__device__ __forceinline__ void stage_b128(const float* g, float* l, bool pred) {
#if USE_ASYNC_ASM
    if (pred) {
        async_ld_b128(g, l);             // ASYNCcnt-tracked, per-lane EXEC
    } else {
        v4f z = {};
        *(v4f*)l = z;                    // tail rows -> zeros (DScnt, barrier)
    }
#else
    v4f v = {};
    if (pred) v = *(const v4f*)g;
    *(v4f*)l = v;
#endif
}

// ---------------------------------------------------------------------------
// Kernel 1: q_hyp = tanh-map(query @ W^T); also q2[b] = tanh^2(||q_proj_b||)
// ---------------------------------------------------------------------------
__global__ __launch_bounds__(256) void project_kernel(
    const float* __restrict__ query, const float* __restrict__ W,
    float* __restrict__ qhyp, float* __restrict__ q2)
{
    const int b = blockIdx.x;
    const int t = threadIdx.x;
    __shared__ float qrow[DIM];
    qrow[t] = query[b * DIM + t];
    __syncthreads();

    const float* wrow = W + (size_t)t * DIM;
    float acc = 0.0f;
    #pragma unroll 8
    for (int k = 0; k < DIM; ++k) acc = fmaf(qrow[k], wrow[k], acc);

    __shared__ float red[256];
    red[t] = acc * acc;
    __syncthreads();
    for (int off = 128; off > 0; off >>= 1) {
        if (t < off) red[t] += red[t + off];
        __syncthreads();
    }
    __shared__ float sc_s, q2_s;
    if (t == 0) {
        float n  = sqrtf(red[0]);
        float th = tanhf(n);
        sc_s = th / fmaxf(n, 1e-6f);
        q2_s = th * th;
    }
    __syncthreads();
    qhyp[b * DIM + t] = sc_s * acc;
    if (t == 0) q2[b] = q2_s;
}

// ---------------------------------------------------------------------------
// Kernel 2: per memory row j: s = tanh(n)/max(n,eps), m2 = tanh^2(n),
//           pen = tanh(n)*(1-decay[j]).  One wave32 per row.
// ---------------------------------------------------------------------------
__global__ __launch_bounds__(256) void stats_kernel(
    const float* __restrict__ mt, const float* __restrict__ decay,
    float* __restrict__ stats, int N)
{
    const int w    = threadIdx.x >> 5;
    const int lane = threadIdx.x & 31;
    const int row  = blockIdx.x * 8 + w;
    if (row >= N) return;

    const float* x = mt + (size_t)row * DIM;
    v4f a = *(const v4f*)(x + lane * 4);
    v4f c = *(const v4f*)(x + 128 + lane * 4);
    float ss = a.x*a.x + a.y*a.y + a.z*a.z + a.w*a.w
             + c.x*c.x + c.y*c.y + c.z*c.z + c.w*c.w;
    #pragma unroll
    for (int m = 16; m > 0; m >>= 1) ss += __shfl_xor(ss, m, 32);

    if (lane == 0) {
        float n  = sqrtf(ss);
        float th = tanhf(n);
        v4f st;
        st.x = th / fmaxf(n, 1e-6f);       // s_j
        st.y = th * th;                    // ||mem_j||^2
        st.z = th * (1.0f - decay[row]);   // penalty
        st.w = 0.0f;
        *(v4f*)(stats + (size_t)row * 4) = st;
    }
}

// ---------------------------------------------------------------------------
// Kernel 3: WMMA f32 GEMM  dot_raw[b][n] = <q_hyp_b, x_n>  + score epilogue.
// WG = 256 threads (8 wave32), WG tile = 128 N-columns, full B=64.
// K chunked by 32 into DOUBLE-BUFFERED LDS filled by async-to-LDS copies:
//   chunk c+1 streams in (ASYNCcnt) while WMMAs consume chunk c.
// LDS row stride 36 floats: 16B-aligned rows, banks (4r+kk) mod 64 distinct.
// ---------------------------------------------------------------------------
__global__ __launch_bounds__(256) void score_kernel(
    const float* __restrict__ mt,   const float* __restrict__ qhyp,
    const float* __restrict__ q2,   const float* __restrict__ stats,
    float* __restrict__ scores, int N)
{
    __shared__ float lds_m[2][128 * 36];   // 2 x 18432 B
    __shared__ float lds_q[2][ 64 * 36];   // 2 x  9216 B

    const int t    = threadIdx.x;
    const int lane = t & 31;
    const int w    = t >> 5;
    const int row0 = blockIdx.x * 128;
    const bool full = (row0 + 128 <= N);   // WG-uniform: all rows valid

    v8f acc0 = {}, acc1 = {}, acc2 = {}, acc3 = {};

    const int koff = (lane < 16) ? 0 : 2;   // K-pair select (ISA A/B layout)
    const int nrow = w * 16 + (lane & 15);  // this lane's column in WG tile
    const int mrow = (lane & 15);

    // per-thread staging coordinates (fixed across chunks)
    // m-tile: 128 rows x 32 K = 1024 x b128 -> 4 per thread
    // q-tile:  64 rows x 32 K =  512 x b128 -> 2 per thread
    int m_r[4], m_kv[4], q_r[2], q_kv[2];
    #pragma unroll
    for (int i = 0; i < 4; ++i) {
        int f = t + i * 256;
        m_r[i]  = f >> 3;          // 0..127
        m_kv[i] = (f & 7) << 2;    // 0..28
    }
    #pragma unroll
    for (int i = 0; i < 2; ++i) {
        int f = t + i * 256;
        q_r[i]  = f >> 3;          // 0..63
        q_kv[i] = (f & 7) << 2;
    }

    // ---- stage one K-chunk (kc = chunk*32) into buffer `buf` -------------
    auto stage_chunk = [&](int chunk, int buf) {
        const int kc = chunk * 32;
        #pragma unroll
        for (int i = 0; i < 4; ++i) {
            int gr = row0 + m_r[i];
            stage_b128(mt + (size_t)gr * DIM + kc + m_kv[i],
                       &lds_m[buf][m_r[i] * 36 + m_kv[i]], gr < N);
        }
        #pragma unroll
        for (int i = 0; i < 2; ++i) {
            stage_b128(qhyp + (size_t)q_r[i] * DIM + kc + q_kv[i],
                       &lds_q[buf][q_r[i] * 36 + q_kv[i]], true);
        }
    };

    stage_chunk(0, 0);                       // prologue: chunk 0 -> buf 0
    for (int c = 0; c < 8; ++c) {
        const int buf = c & 1;
        const bool have_next = (c + 1 < 8);
        if (have_next) stage_chunk(c + 1, buf ^ 1);   // prefetch next chunk
#if USE_ASYNC_ASM
        // chunk c's 6 async ops complete before the 6 just issued (in-order)
        if (have_next && full) wait_async_le6();
        else                   wait_async_0();
#endif
        __syncthreads();                     // all waves' chunk-c data visible

        #pragma unroll
        for (int kk = 0; kk < 8; ++kk) {
            const int off = kk * 4 + koff;
            v2f bfrag = *(const v2f*)(&lds_m[buf][nrow * 36 + off]);
            v2f a0 = *(const v2f*)(&lds_q[buf][(0  + mrow) * 36 + off]);
            v2f a1 = *(const v2f*)(&lds_q[buf][(16 + mrow) * 36 + off]);
            v2f a2 = *(const v2f*)(&lds_q[buf][(32 + mrow) * 36 + off]);
            v2f a3 = *(const v2f*)(&lds_q[buf][(48 + mrow) * 36 + off]);
            acc0 = __builtin_amdgcn_wmma_f32_16x16x4_f32(false, a0, false, bfrag, (short)0, acc0, false, false);
            acc1 = __builtin_amdgcn_wmma_f32_16x16x4_f32(false, a1, false, bfrag, (short)0, acc1, false, false);
            acc2 = __builtin_amdgcn_wmma_f32_16x16x4_f32(false, a2, false, bfrag, (short)0, acc2, false, false);
            acc3 = __builtin_amdgcn_wmma_f32_16x16x4_f32(false, a3, false, bfrag, (short)0, acc3, false, false);
        }
        __syncthreads();   // buf[c&1] free before chunk c+2 streams into it
    }

    // Epilogue: C/D layout -> element i of v8f is M = mt*16 + i + (lane<16?0:8),
    // N = lane%16 within the wave's 16-column slab.
    const int n = row0 + w * 16 + (lane & 15);
    if (n < N) {
        v4f st = *(const v4f*)(stats + (size_t)n * 4);
        const float s = st.x, m2 = st.y, pen = st.z;
        const float one_m2 = 1.0f - m2;
        const int mhalf = (lane < 16) ? 0 : 8;
        v8f accs[4] = {acc0, acc1, acc2, acc3};
        #pragma unroll
        for (int mt_ = 0; mt_ < 4; ++mt_) {
            #pragma unroll
            for (int i = 0; i < 8; ++i) {
                int   m   = mt_ * 16 + i + mhalf;
                float dot = s * accs[mt_][i];
                float q2v = q2[m];
                float sq    = fmaxf(q2v + m2 - 2.0f * dot, 0.0f);
                float denom = fmaxf((1.0f - q2v) * one_m2, 1e-9f);
                float arg   = fmaxf(1.0f + 2.0f * sq / denom, 1.0f + 1e-7f);
                float dist  = logf(arg + sqrtf(arg * arg - 1.0f));   // arccosh
                scores[(size_t)m * N + n] = -(dist + pen);
            }
        }
    }
}

// ---------------------------------------------------------------------------
// Kernel 4: top-32 per query via 32 masked argmax sweeps (tie -> lowest index).
// ---------------------------------------------------------------------------
__global__ __launch_bounds__(256) void topk_kernel(
    float* __restrict__ scores, int N, int K,
    int* __restrict__ idx_ws, float* __restrict__ out_idx_f)
{
    const int b = blockIdx.x;
    float* row = scores + (size_t)b * N;
    __shared__ float ss[256];
    __shared__ int   si[256];

    for (int k = 0; k < K; ++k) {
        float best = -INFINITY;
        int   bi   = 0x7fffffff;
        for (int i = threadIdx.x; i < N; i += 256) {
            float v = row[i];
            if (v > best || (v == best && i < bi)) { best = v; bi = i; }
        }
        ss[threadIdx.x] = best; si[threadIdx.x] = bi;
        __syncthreads();
        for (int off = 128; off > 0; off >>= 1) {
            if (threadIdx.x < off) {
                float v = ss[threadIdx.x + off];
                int   j = si[threadIdx.x + off];
                if (v > ss[threadIdx.x] || (v == ss[threadIdx.x] && j < si[threadIdx.x])) {
                    ss[threadIdx.x] = v; si[threadIdx.x] = j;
                }
            }
            __syncthreads();
        }
        if (threadIdx.x == 0) {
            int j = si[0];
            idx_ws[b * K + k]    = j;
            out_idx_f[b * K + k] = (float)j;
            row[j] = -INFINITY;
        }
        __syncthreads();
    }
}

// ---------------------------------------------------------------------------
// Kernel 5: retrieved = log-map(tanh-map(x_j)) with the reference's clamp.
// ---------------------------------------------------------------------------
__global__ __launch_bounds__(256) void gather_kernel(
    const float* __restrict__ mt, const int* __restrict__ idx_ws,
    float* __restrict__ out)
{
    const int bk = blockIdx.x;              // 0 .. B*K-1
    const int j  = idx_ws[bk];
    const float* x = mt + (size_t)j * DIM;
    float v = x[threadIdx.x];

    __shared__ float red[256];
    red[threadIdx.x] = v * v;
    __syncthreads();
    for (int off = 128; off > 0; off >>= 1) {
        if (threadIdx.x < off) red[threadIdx.x] += red[threadIdx.x + off];
        __syncthreads();
    }
    __shared__ float coef_s;
    if (threadIdx.x == 0) {
        float n  = sqrtf(red[0]);
        float nm = tanhf(n);                               // ||mem||
        float s  = nm / fmaxf(n, 1e-6f);                   // tanh-map scale
        float nc = fminf(fmaxf(nm, 1e-6f), 1.0f - 1e-5f);
        float at = 0.5f * logf((1.0f + nc) / (1.0f - nc)); // arctanh
        coef_s = at / fmaxf(nm, 1e-6f) * s;
    }
    __syncthreads();
    out[(size_t)bk * DIM + threadIdx.x] = coef_s * v;
}

// ---------------------------------------------------------------------------
extern "C" void kernel_launch(void* const* d_in, const int* in_sizes, int n_in,
                              void* d_out, int out_size, void* d_ws, size_t ws_size,
                              hipStream_t stream)
{
    const float* query = (const float*)d_in[0];   // (64,256)
    const float* W     = (const float*)d_in[1];   // (256,256)
    const float* mt    = (const float*)d_in[2];   // (N,256)
    const float* decay = (const float*)d_in[3];   // (N,)
    const int    N     = in_sizes[3];
    const int    K     = TOPK;                    // top_k = 32 (fixed by setup)

    // workspace layout, offsets 256B-aligned
    char* ws = (char*)d_ws;
    float* qhyp   = (float*)(ws);                                   // 64*256
    float* q2     = (float*)(ws + 65536);                           // 64
    float* stats  = (float*)(ws + 66048);                           // N*4
    int*   idx_ws = (int*)  (ws + 66048 + (size_t)N * 16);          // B*K
    size_t sc_off = 66048 + (size_t)N * 16 + (size_t)BQ * K * 4;
    sc_off = (sc_off + 255) & ~(size_t)255;
    float* scores = (float*)(ws + sc_off);                          // 64*N

    float* out_retr = (float*)d_out;                   // (B,K,256)
    float* out_idx  = out_retr + (size_t)BQ * K * DIM; // (B,K) as float

    project_kernel<<<BQ, 256, 0, stream>>>(query, W, qhyp, q2);
    stats_kernel  <<<(N + 7) / 8, 256, 0, stream>>>(mt, decay, stats, N);
    score_kernel  <<<(N + 127) / 128, 256, 0, stream>>>(mt, qhyp, q2, stats, scores, N);
    topk_kernel   <<<BQ, 256, 0, stream>>>(scores, N, K, idx_ws, out_idx);
    gather_kernel <<<BQ * K, 256, 0, stream>>>(mt, idx_ws, out_retr);
}